// gatGCN_16286515986690
// MI455X (gfx1250) — compile-verified
//
#include <hip/hip_runtime.h>
#include <math.h>

// ---------------------------------------------------------------------------
// GAT 2-layer forward for MI455X (gfx1250, wave32, WMMA).
// GEMMs use V_WMMA_F32_16X16X4_F32 (fp32 in/out -> no precision loss vs ref).
// Edge softmax/aggregation: 3-pass segment softmax with f32 atomics (L2-res).
// ---------------------------------------------------------------------------

typedef __attribute__((ext_vector_type(2))) float v2f;
typedef __attribute__((ext_vector_type(8))) float v8f;

#define NEG_SLOPE 0.2f

// Sign-aware float atomic max via integer ordering trick (always compiles,
// correct for all finite floats; m initialized to -1e30).
__device__ inline void atomicMaxF(float* addr, float val) {
  if (val >= 0.0f) {
    atomicMax((int*)addr, __float_as_int(val));
  } else {
    atomicMin((unsigned int*)addr, __float_as_uint(val));
  }
}

__device__ inline void edge_sd(const int* __restrict__ ei, int Ee, int e,
                               int& s, int& d) {
  if (e < Ee) { s = ei[e]; d = ei[Ee + e]; }
  else        { s = e - Ee; d = e - Ee; }   // self-loops appended
}

// ---------------------------------------------------------------------------
// GEMM1: h1[N,64] = x[N,128] @ W1[128,64]   (WMMA f32 16x16x4)
// grid.x = N/16 tiles, blockDim = (32,4): wave ty handles output cols ty*16..+15
// ---------------------------------------------------------------------------
__global__ __launch_bounds__(128)
void k_gemm1(const float* __restrict__ x, const float* __restrict__ W,
             float* __restrict__ h1, int Nn) {
  const int lane = threadIdx.x & 31;
  const int half = lane >> 4;          // selects K pair {0,1} vs {2,3}
  const int r16  = lane & 15;
  const int mtile = blockIdx.x;
  int row = mtile * 16 + r16;
  if (row >= Nn) row = Nn - 1;         // clamp (stores guarded below)
  const int ncol = threadIdx.y * 16 + r16;

  v8f acc = {};
  #pragma unroll
  for (int k = 0; k < 128; k += 4) {
    const int k0 = k + 2 * half;
    v2f a, b;
    a.x = x[row * 128 + k0];
    a.y = x[row * 128 + k0 + 1];
    b.x = W[k0 * 64 + ncol];
    b.y = W[(k0 + 1) * 64 + ncol];
    acc = __builtin_amdgcn_wmma_f32_16x16x4_f32(false, a, false, b,
                                                (short)0, acc, false, false);
  }
  const int mbase = mtile * 16 + 8 * half;
  #pragma unroll
  for (int v = 0; v < 8; ++v) {
    const int m = mbase + v;
    if (m < Nn) h1[m * 64 + ncol] = acc[v];
  }
}

// ---------------------------------------------------------------------------
// GEMM2: h2[N,16] = h1act[N,64] @ W2[64,16]   (WMMA f32 16x16x4), 1 wave/tile
// ---------------------------------------------------------------------------
__global__ __launch_bounds__(32)
void k_gemm2(const float* __restrict__ in, const float* __restrict__ W,
             float* __restrict__ h2, int Nn) {
  const int lane = threadIdx.x & 31;
  const int half = lane >> 4;
  const int r16  = lane & 15;
  const int mtile = blockIdx.x;
  int row = mtile * 16 + r16;
  if (row >= Nn) row = Nn - 1;
  const int ncol = r16;

  v8f acc = {};
  #pragma unroll
  for (int k = 0; k < 64; k += 4) {
    const int k0 = k + 2 * half;
    v2f a, b;
    a.x = in[row * 64 + k0];
    a.y = in[row * 64 + k0 + 1];
    b.x = W[k0 * 16 + ncol];
    b.y = W[(k0 + 1) * 16 + ncol];
    acc = __builtin_amdgcn_wmma_f32_16x16x4_f32(false, a, false, b,
                                                (short)0, acc, false, false);
  }
  const int mbase = mtile * 16 + 8 * half;
  #pragma unroll
  for (int v = 0; v < 8; ++v) {
    const int m = mbase + v;
    if (m < Nn) h2[m * 16 + ncol] = acc[v];
  }
}

// ---------------------------------------------------------------------------
// Layer-1 per-node attention logits; also initializes m/s/o for the edge phase
// one thread per (node, head)
// ---------------------------------------------------------------------------
__global__ void k_alpha1(const float* __restrict__ h1,
                         const float* __restrict__ asrc,
                         const float* __restrict__ adst,
                         float* __restrict__ as1, float* __restrict__ ad1,
                         float* __restrict__ m1, float* __restrict__ s1,
                         float* __restrict__ o1, int Nn) {
  const int idx = blockIdx.x * blockDim.x + threadIdx.x;
  if (idx >= Nn * 8) return;
  const int n = idx >> 3;
  const int h = idx & 7;
  float sa = 0.0f, sd = 0.0f;
  #pragma unroll
  for (int c = 0; c < 8; ++c) {
    const float hv = h1[n * 64 + h * 8 + c];
    sa += hv * asrc[h * 8 + c];
    sd += hv * adst[h * 8 + c];
    o1[n * 64 + h * 8 + c] = 0.0f;
  }
  as1[idx] = sa;
  ad1[idx] = sd;
  m1[idx]  = -1e30f;
  s1[idx]  = 0.0f;
}

__global__ void k_edge_max1(const int* __restrict__ ei, int Ee, int Nn,
                            const float* __restrict__ as1,
                            const float* __restrict__ ad1,
                            float* __restrict__ m1) {
  const int e = blockIdx.x * blockDim.x + threadIdx.x;
  if (e >= Ee + Nn) return;
  int s, d; edge_sd(ei, Ee, e, s, d);
  #pragma unroll
  for (int h = 0; h < 8; ++h) {
    float v = as1[s * 8 + h] + ad1[d * 8 + h];
    v = (v > 0.0f) ? v : NEG_SLOPE * v;
    atomicMaxF(&m1[d * 8 + h], v);
  }
}

__global__ void k_edge_sum1(const int* __restrict__ ei, int Ee, int Nn,
                            const float* __restrict__ as1,
                            const float* __restrict__ ad1,
                            const float* __restrict__ m1,
                            float* __restrict__ s1) {
  const int e = blockIdx.x * blockDim.x + threadIdx.x;
  if (e >= Ee + Nn) return;
  int s, d; edge_sd(ei, Ee, e, s, d);
  #pragma unroll
  for (int h = 0; h < 8; ++h) {
    float v = as1[s * 8 + h] + ad1[d * 8 + h];
    v = (v > 0.0f) ? v : NEG_SLOPE * v;
    atomicAdd(&s1[d * 8 + h], expf(v - m1[d * 8 + h]));
  }
}

// one thread per (edge, head): 8 feature atomics each
__global__ void k_edge_agg1(const int* __restrict__ ei, int Ee, int Nn,
                            const float* __restrict__ as1,
                            const float* __restrict__ ad1,
                            const float* __restrict__ m1,
                            const float* __restrict__ s1,
                            const float* __restrict__ h1,
                            float* __restrict__ o1) {
  const int tid = blockIdx.x * blockDim.x + threadIdx.x;
  if (tid >= (Ee + Nn) * 8) return;
  const int e = tid >> 3;
  const int h = tid & 7;
  int s, d; edge_sd(ei, Ee, e, s, d);
  float v = as1[s * 8 + h] + ad1[d * 8 + h];
  v = (v > 0.0f) ? v : NEG_SLOPE * v;
  const float alpha = expf(v - m1[d * 8 + h]) / (s1[d * 8 + h] + 1e-16f);
  const float* hs = &h1[s * 64 + h * 8];
  float* od = &o1[d * 64 + h * 8];
  #pragma unroll
  for (int c = 0; c < 8; ++c) atomicAdd(&od[c], hs[c] * alpha);
}

// bias + ELU in place
__global__ void k_elu1(float* __restrict__ o1, const float* __restrict__ b1,
                       int Nn) {
  const int idx = blockIdx.x * blockDim.x + threadIdx.x;
  if (idx >= Nn * 64) return;
  const float v = o1[idx] + b1[idx & 63];
  o1[idx] = (v > 0.0f) ? v : expm1f(v);
}

// ---------------------------------------------------------------------------
// Layer-2 per-node logits (H=1, C=16); folds init of m/s/o
// ---------------------------------------------------------------------------
__global__ void k_alpha2(const float* __restrict__ h2,
                         const float* __restrict__ asrc,
                         const float* __restrict__ adst,
                         float* __restrict__ as2, float* __restrict__ ad2,
                         float* __restrict__ m2, float* __restrict__ s2,
                         float* __restrict__ o2, int Nn) {
  const int n = blockIdx.x * blockDim.x + threadIdx.x;
  if (n >= Nn) return;
  float sa = 0.0f, sd = 0.0f;
  #pragma unroll
  for (int c = 0; c < 16; ++c) {
    const float hv = h2[n * 16 + c];
    sa += hv * asrc[c];
    sd += hv * adst[c];
    o2[n * 16 + c] = 0.0f;
  }
  as2[n] = sa;
  ad2[n] = sd;
  m2[n]  = -1e30f;
  s2[n]  = 0.0f;
}

__global__ void k_edge_max2(const int* __restrict__ ei, int Ee, int Nn,
                            const float* __restrict__ as2,
                            const float* __restrict__ ad2,
                            float* __restrict__ m2) {
  const int e = blockIdx.x * blockDim.x + threadIdx.x;
  if (e >= Ee + Nn) return;
  int s, d; edge_sd(ei, Ee, e, s, d);
  float v = as2[s] + ad2[d];
  v = (v > 0.0f) ? v : NEG_SLOPE * v;
  atomicMaxF(&m2[d], v);
}

__global__ void k_edge_sum2(const int* __restrict__ ei, int Ee, int Nn,
                            const float* __restrict__ as2,
                            const float* __restrict__ ad2,
                            const float* __restrict__ m2,
                            float* __restrict__ s2) {
  const int e = blockIdx.x * blockDim.x + threadIdx.x;
  if (e >= Ee + Nn) return;
  int s, d; edge_sd(ei, Ee, e, s, d);
  float v = as2[s] + ad2[d];
  v = (v > 0.0f) ? v : NEG_SLOPE * v;
  atomicAdd(&s2[d], expf(v - m2[d]));
}

__global__ void k_edge_agg2(const int* __restrict__ ei, int Ee, int Nn,
                            const float* __restrict__ as2,
                            const float* __restrict__ ad2,
                            const float* __restrict__ m2,
                            const float* __restrict__ s2,
                            const float* __restrict__ h2,
                            float* __restrict__ o2) {
  const int e = blockIdx.x * blockDim.x + threadIdx.x;
  if (e >= Ee + Nn) return;
  int s, d; edge_sd(ei, Ee, e, s, d);
  float v = as2[s] + ad2[d];
  v = (v > 0.0f) ? v : NEG_SLOPE * v;
  const float alpha = expf(v - m2[d]) / (s2[d] + 1e-16f);
  const float* hs = &h2[s * 16];
  float* od = &o2[d * 16];
  #pragma unroll
  for (int c = 0; c < 16; ++c) atomicAdd(&od[c], hs[c] * alpha);
}

// bias + log_softmax -> d_out
__global__ void k_out(const float* __restrict__ o2, const float* __restrict__ b2,
                      float* __restrict__ out, int Nn) {
  const int n = blockIdx.x * blockDim.x + threadIdx.x;
  if (n >= Nn) return;
  float row[16];
  float mx = -1e30f;
  #pragma unroll
  for (int c = 0; c < 16; ++c) {
    row[c] = o2[n * 16 + c] + b2[c];
    mx = fmaxf(mx, row[c]);
  }
  float sum = 0.0f;
  #pragma unroll
  for (int c = 0; c < 16; ++c) sum += expf(row[c] - mx);
  const float lse = mx + logf(sum);
  #pragma unroll
  for (int c = 0; c < 16; ++c) out[n * 16 + c] = row[c] - lse;
}

// ---------------------------------------------------------------------------
extern "C" void kernel_launch(void* const* d_in, const int* in_sizes, int n_in,
                              void* d_out, int out_size, void* d_ws, size_t ws_size,
                              hipStream_t stream) {
  const float* x     = (const float*)d_in[0];
  const int*   ei    = (const int*)  d_in[1];   // [2,E]: row0=src, row1=dst
  const float* W1    = (const float*)d_in[2];
  const float* asrc1 = (const float*)d_in[3];
  const float* adst1 = (const float*)d_in[4];
  const float* b1    = (const float*)d_in[5];
  const float* W2    = (const float*)d_in[6];
  const float* asrc2 = (const float*)d_in[7];
  const float* adst2 = (const float*)d_in[8];
  const float* b2    = (const float*)d_in[9];
  float* out = (float*)d_out;

  const int Nn = in_sizes[0] / 128;   // 50000
  const int Ee = in_sizes[1] / 2;     // 800000
  const int EN = Ee + Nn;             // edges incl. self-loops

  // workspace carve-up (fp32): total N*196 floats (~39.2 MB for N=50000)
  float* ws  = (float*)d_ws;
  float* h1  = ws;              // [N,64]
  float* o1  = h1  + (size_t)Nn * 64;  // [N,64]
  float* as1 = o1  + (size_t)Nn * 64;  // [N,8]
  float* ad1 = as1 + (size_t)Nn * 8;
  float* m1  = ad1 + (size_t)Nn * 8;
  float* s1w = m1  + (size_t)Nn * 8;
  float* h2  = s1w + (size_t)Nn * 8;   // [N,16]
  float* o2  = h2  + (size_t)Nn * 16;  // [N,16]
  float* as2 = o2  + (size_t)Nn * 16;  // [N]
  float* ad2 = as2 + (size_t)Nn;
  float* m2  = ad2 + (size_t)Nn;
  float* s2w = m2  + (size_t)Nn;

  const int T = 256;
  const int mtiles = (Nn + 15) / 16;

  // ---- Layer 1 ----
  k_gemm1<<<dim3(mtiles), dim3(32, 4), 0, stream>>>(x, W1, h1, Nn);
  k_alpha1<<<(Nn * 8 + T - 1) / T, T, 0, stream>>>(h1, asrc1, adst1,
                                                   as1, ad1, m1, s1w, o1, Nn);
  k_edge_max1<<<(EN + T - 1) / T, T, 0, stream>>>(ei, Ee, Nn, as1, ad1, m1);
  k_edge_sum1<<<(EN + T - 1) / T, T, 0, stream>>>(ei, Ee, Nn, as1, ad1, m1, s1w);
  k_edge_agg1<<<((size_t)EN * 8 + T - 1) / T, T, 0, stream>>>(
      ei, Ee, Nn, as1, ad1, m1, s1w, h1, o1);
  k_elu1<<<(Nn * 64 + T - 1) / T, T, 0, stream>>>(o1, b1, Nn);

  // ---- Layer 2 ----
  k_gemm2<<<dim3(mtiles), dim3(32), 0, stream>>>(o1, W2, h2, Nn);
  k_alpha2<<<(Nn + T - 1) / T, T, 0, stream>>>(h2, asrc2, adst2,
                                               as2, ad2, m2, s2w, o2, Nn);
  k_edge_max2<<<(EN + T - 1) / T, T, 0, stream>>>(ei, Ee, Nn, as2, ad2, m2);
  k_edge_sum2<<<(EN + T - 1) / T, T, 0, stream>>>(ei, Ee, Nn, as2, ad2, m2, s2w);
  k_edge_agg2<<<(EN + T - 1) / T, T, 0, stream>>>(ei, Ee, Nn, as2, ad2, m2, s2w,
                                                  h2, o2);
  k_out<<<(Nn + T - 1) / T, T, 0, stream>>>(o2, b2, out, Nn);
}